// HeteroRGCNLayer_12506944766357
// MI455X (gfx1250) — compile-verified
//
#include <hip/hip_runtime.h>
#include <hip/hip_bf16.h>
#include <math.h>
#include <stdint.h>

// ---------------------------------------------------------------------------
// Sizes (match reference)
// ---------------------------------------------------------------------------
#define D        256
#define N_TOK    50000
#define N_SRL    20000
#define N_ENT    10000
#define T_ROWS   50000
#define E_TS     300000
#define E_EE     150000
#define E_ST     300000
#define E_ET     150000
#define SLOPE    0.01f
#define CHROWS   512
#define NCH      ((T_ROWS + CHROWS - 1) / CHROWS)   // 98

typedef float v2f __attribute__((ext_vector_type(2)));
typedef float v8f __attribute__((ext_vector_type(8)));
typedef int   v4i __attribute__((ext_vector_type(4)));

#if __has_builtin(__builtin_amdgcn_global_load_async_to_lds_b128)
#define HAS_ASYNC 1
#endif

typedef __attribute__((address_space(1))) v4i* gv4p;   // global int4*
typedef __attribute__((address_space(3))) v4i* lv4p;   // LDS int4*

// 16-byte global->LDS copy: async (CDNA5 path, ASYNCcnt) or sync fallback
__device__ __forceinline__ void cp16(const float* g, float* l) {
#ifdef HAS_ASYNC
    __builtin_amdgcn_global_load_async_to_lds_b128(
        (gv4p)(uintptr_t)g, (lv4p)(uint32_t)(uintptr_t)l, 0, 0);
#else
    *(float4*)l = *(const float4*)g;
#endif
}

__device__ __forceinline__ void wait_async() {
#ifdef HAS_ASYNC
#if __has_builtin(__builtin_amdgcn_s_wait_asynccnt)
    __builtin_amdgcn_s_wait_asynccnt(0);
#else
    asm volatile("s_wait_asynccnt 0x0" ::: "memory");
#endif
#endif
}

// ---------------------------------------------------------------------------
// WMMA fp32 GEMM:  C[M,N] = A[M,K] @ W[:, wofs:wofs+K]^T (+bias)
//   A: row-major M x K (lda = K), K multiple of 64 (always 256 here)
//   W: row-major N x ldw, slice columns [wofs, wofs+K)
//   N multiple of 64.  Block = 128 threads (4 waves), 64x64 C tile.
//   Double-buffered LDS tiles fed by async global->LDS copies.
// ---------------------------------------------------------------------------
#define TK 64
#define LDS_STRIDE 68   // pad: keeps float4/b128 16B-aligned, avoids bank conflicts

__global__ __launch_bounds__(128)
void wmma_gemm_bias(const float* __restrict__ A, int M, int K,
                    const float* __restrict__ W, int ldw, int wofs,
                    const float* __restrict__ bias,
                    float* __restrict__ C, int ldc)
{
    __shared__ float sA[2][64][LDS_STRIDE];
    __shared__ float sB[2][64][LDS_STRIDE];

    const int m0   = blockIdx.x * 64;
    const int n0   = blockIdx.y * 64;
    const int tid  = threadIdx.x;
    const int wave = tid >> 5;
    const int lane = tid & 31;

    // pre-zero out-of-range A rows in BOTH buffers (async copies skip them,
    // so the zeros persist across every K chunk)
    if (m0 + 64 > M) {
        float4 z = make_float4(0.f, 0.f, 0.f, 0.f);
        for (int i = tid; i < 1024; i += 128) {
            int r = i >> 4, c4 = (i & 15) << 2;
            if (m0 + r >= M) {
                *(float4*)&sA[0][r][c4] = z;
                *(float4*)&sA[1][r][c4] = z;
            }
        }
    }

    // tile copy: each wave issues 16 async b128 ops (8 A + 8 B)
    auto copy_tile = [&](int buf, int k0) {
        for (int i = tid; i < 1024; i += 128) {
            int r  = i >> 4;
            int c4 = (i & 15) << 2;
            int gr = m0 + r;
            if (gr < M)
                cp16(A + (size_t)gr * K + k0 + c4, &sA[buf][r][c4]);
            cp16(W + (size_t)(n0 + r) * ldw + wofs + k0 + c4, &sB[buf][r][c4]);
        }
    };

    v8f acc[4] = {};

    const int nk = K / TK;
    copy_tile(0, 0);
    wait_async();
    __syncthreads();

    for (int kc = 0; kc < nk; ++kc) {
        if (kc + 1 < nk) copy_tile((kc + 1) & 1, (kc + 1) * TK);

        const int buf   = kc & 1;
        const int r0    = wave * 16;
        const int lrow  = lane & 15;
        const int ksel  = (lane >> 4) << 1;   // lanes 0-15: K+0,1 ; 16-31: K+2,3
        #pragma unroll 4
        for (int k = 0; k < TK; k += 4) {
            int kk = k + ksel;
            v2f af = *(const v2f*)&sA[buf][r0 + lrow][kk];
            #pragma unroll
            for (int j = 0; j < 4; ++j) {
                v2f bf = *(const v2f*)&sB[buf][j * 16 + lrow][kk];
                acc[j] = __builtin_amdgcn_wmma_f32_16x16x4_f32(
                    false, af, false, bf, (short)0, acc[j], false, false);
            }
        }
        wait_async();
        __syncthreads();
    }

    // C/D layout: VGPR v -> row base+v (lanes 0-15) / base+8+v (lanes 16-31)
    const int rbase = m0 + wave * 16 + ((lane >> 4) << 3);
    const int lrow  = lane & 15;
    #pragma unroll
    for (int j = 0; j < 4; ++j) {
        int c = n0 + j * 16 + lrow;
        float bv = bias ? bias[c] : 0.f;
        #pragma unroll
        for (int v = 0; v < 8; ++v) {
            int r = rbase + v;
            if (r < M) C[(size_t)r * ldc + c] = acc[j][v] + bv;
        }
    }
}

// ---------------------------------------------------------------------------
// Small helpers
// ---------------------------------------------------------------------------
__global__ void zero_kernel(float* p, int n) {
    int i = blockIdx.x * 256 + threadIdx.x;
    if (i < n) p[i] = 0.f;
}
__global__ void fill_u32(unsigned* p, unsigned v, int n) {
    int i = blockIdx.x * 256 + threadIdx.x;
    if (i < n) p[i] = v;
}

__device__ __forceinline__ float leaky(float x) { return x >= 0.f ? x : SLOPE * x; }

// monotonic float<->uint mapping for atomicMax on floats
__device__ __forceinline__ unsigned fkey(float x) {
    int i = __float_as_int(x);
    return (i >= 0) ? ((unsigned)i | 0x80000000u) : ~(unsigned)i;
}
__device__ __forceinline__ float fdecode(unsigned k) {
    int i = (k & 0x80000000u) ? (int)(k & 0x7FFFFFFFu) : (int)~k;
    return __int_as_float(i);
}

// ---------------------------------------------------------------------------
// Segment-softmax attention (3 atomic passes), D threads per edge-block.
// edge_level=1: asrc/vals indexed by edge; else gathered via src[e].
// ---------------------------------------------------------------------------
__device__ __forceinline__ float att_e(const float* asrc, const int* src,
                                       const float* adst, int e, int d, int f,
                                       int edge_level) {
    float s = edge_level ? asrc[(size_t)e * D + f] : asrc[(size_t)src[e] * D + f];
    return leaky(s + adst[(size_t)d * D + f]);
}

__global__ void att_pass_max(const float* asrc, const int* src,
                             const float* adst, const int* dst,
                             unsigned* umax, int edge_level) {
    int e = blockIdx.x, f = threadIdx.x;
    int d = dst[e];
    float ev = att_e(asrc, src, adst, e, d, f, edge_level);
    atomicMax(&umax[(size_t)d * D + f], fkey(ev));
}

// decode max (nonfinite -> 0), zero den and h accumulators (all nseg*D)
__global__ void att_prep(unsigned* umax, float* den, float* h, int n) {
    int i = blockIdx.x * 256 + threadIdx.x;
    if (i >= n) return;
    float m = fdecode(umax[i]);
    if (!isfinite(m)) m = 0.f;
    ((float*)umax)[i] = m;
    den[i] = 0.f;
    h[i]   = 0.f;
}

__global__ void att_pass_den(const float* asrc, const int* src,
                             const float* adst, const int* dst,
                             const float* fmax, float* den, int edge_level) {
    int e = blockIdx.x, f = threadIdx.x;
    int d = dst[e];
    float ev = att_e(asrc, src, adst, e, d, f, edge_level);
    atomicAdd(&den[(size_t)d * D + f], expf(ev - fmax[(size_t)d * D + f]));
}

__global__ void att_pass_agg(const float* asrc, const int* src,
                             const float* adst, const int* dst,
                             const float* fmax, const float* den,
                             const float* vals, float* h, int edge_level) {
    int e = blockIdx.x, f = threadIdx.x;
    int d = dst[e];
    size_t o = (size_t)d * D + f;
    float ev    = att_e(asrc, src, adst, e, d, f, edge_level);
    float alpha = expf(ev - fmax[o]) / fmaxf(den[o], 1e-9f);
    float v = edge_level ? vals[(size_t)e * D + f] : vals[(size_t)src[e] * D + f];
    atomicAdd(&h[o], alpha * v);
}

// ---------------------------------------------------------------------------
// Prefix sum of bert_token_emb -> P[(T+1) x D]  (exclusive per row)
// ---------------------------------------------------------------------------
__global__ void csum_partial(const float* x, float* cs) {
    int ch = blockIdx.x, f = threadIdx.x;
    int r0 = ch * CHROWS;
    float s = 0.f;
    for (int i = 0; i < CHROWS; ++i) {
        int r = r0 + i;
        if (r < T_ROWS) s += x[(size_t)r * D + f];
    }
    cs[ch * D + f] = s;
}
__global__ void csum_scan(float* cs) {
    int f = threadIdx.x;
    float run = 0.f;
    for (int ch = 0; ch < NCH; ++ch) {
        float t = cs[ch * D + f];
        cs[ch * D + f] = run;
        run += t;
    }
}
__global__ void csum_write(const float* x, const float* cs, float* P) {
    int ch = blockIdx.x, f = threadIdx.x;
    int r0 = ch * CHROWS;
    float run = cs[ch * D + f];
    for (int i = 0; i < CHROWS; ++i) {
        int r = r0 + i;
        if (r < T_ROWS) {
            P[(size_t)r * D + f] = run;
            run += x[(size_t)r * D + f];
        }
    }
    if (ch == NCH - 1) P[(size_t)T_ROWS * D + f] = run;
}

__global__ void span_rel(const float* P, const int* span, const float* rel_type,
                         float* rel_emb) {
    int e = blockIdx.x, f = threadIdx.x;
    int x = span[2 * e], y = span[2 * e + 1];
    float sm = (P[(size_t)y * D + f] - P[(size_t)x * D + f]) / (float)(y - x);
    rel_emb[(size_t)e * D + f] = rel_type[e] * sm;
}

__global__ void edge_add_gather(float* m, const float* node, const int* src) {
    int e = blockIdx.x, f = threadIdx.x;
    m[(size_t)e * D + f] += node[(size_t)src[e] * D + f];
}

// Wrc[n][k] = sum_j W_common[n][j] * W_rel_trans[j][256+k]
__global__ void wrc_gemm(const float* Wc, const float* Wrel, float* Wrc) {
    int n = blockIdx.x, k = threadIdx.x;
    float s = 0.f;
    for (int j = 0; j < D; ++j)
        s += Wc[n * D + j] * Wrel[j * (2 * D) + D + k];
    Wrc[n * D + k] = s;
}

// ---------------------------------------------------------------------------
// GRU
// ---------------------------------------------------------------------------
__device__ __forceinline__ float gru_f(const float* gi, const float* gh, float h, int f) {
    float r = 1.f / (1.f + expf(-(gi[f]       + gh[f])));
    float z = 1.f / (1.f + expf(-(gi[D + f]   + gh[D + f])));
    float n = tanhf(gi[2 * D + f] + r * gh[2 * D + f]);
    return (1.f - z) * n + z * h;
}

__global__ void gru_edge_agg(const float* gi, const int* src, const float* gh,
                             const float* htok, const int* dst, float* out) {
    int e = blockIdx.x, f = threadIdx.x;
    int s = src[e], d = dst[e];
    float m = gru_f(gi + (size_t)s * 3 * D, gh + (size_t)d * 3 * D,
                    htok[(size_t)d * D + f], f);
    atomicAdd(&out[(size_t)d * D + f], m);
}

__global__ void gru_combine(const float* gi, const float* gh, const float* hprev,
                            float* out) {
    int i = blockIdx.x, f = threadIdx.x;
    out[(size_t)i * D + f] = gru_f(gi + (size_t)i * 3 * D, gh + (size_t)i * 3 * D,
                                   hprev[(size_t)i * D + f], f);
}

// ---------------------------------------------------------------------------
// Launch
// ---------------------------------------------------------------------------
static inline void gemm(const float* A, int M, int K, const float* W, int ldw,
                        int wofs, const float* bias, float* C, int N,
                        hipStream_t s) {
    dim3 g((M + 63) / 64, N / 64), b(128);
    hipLaunchKernelGGL(wmma_gemm_bias, g, b, 0, s, A, M, K, W, ldw, wofs, bias, C, N);
}

extern "C" void kernel_launch(void* const* d_in, const int* in_sizes, int n_in,
                              void* d_out, int out_size, void* d_ws, size_t ws_size,
                              hipStream_t stream) {
    const float* feat_tok = (const float*)d_in[0];
    const float* feat_srl = (const float*)d_in[1];
    const float* feat_ent = (const float*)d_in[2];
    const float* bert     = (const float*)d_in[3];
    const float* rel_type = (const float*)d_in[4];
    const int*   src_ts   = (const int*)d_in[5];
    const int*   dst_ts   = (const int*)d_in[6];
    const int*   src_ee   = (const int*)d_in[7];
    const int*   dst_ee   = (const int*)d_in[8];
    const int*   src_st   = (const int*)d_in[9];
    const int*   dst_st   = (const int*)d_in[10];
    const int*   src_et   = (const int*)d_in[11];
    const int*   dst_et   = (const int*)d_in[12];
    const int*   span_idx = (const int*)d_in[13];
    const float* W_nt     = (const float*)d_in[14];
    const float* b_nt     = (const float*)d_in[15];
    const float* W_att    = (const float*)d_in[16];
    const float* b_att    = (const float*)d_in[17];
    const float* W_rel    = (const float*)d_in[18];
    const float* b_rel    = (const float*)d_in[19];
    const float* W_com    = (const float*)d_in[20];
    const float* b_com    = (const float*)d_in[21];
    const float* W_ih     = (const float*)d_in[22];
    const float* W_hh     = (const float*)d_in[23];
    const float* b_ih     = (const float*)d_in[24];
    const float* b_hh     = (const float*)d_in[25];

    float* out_htok = (float*)d_out;
    float* out_hsrl = out_htok + (size_t)N_TOK * D;
    float* out_hent = out_hsrl + (size_t)N_SRL * D;

    // workspace arena (floats); big regions are phase-reused
    float* ws = (float*)d_ws;
    size_t off = 0;
    auto alloc = [&](size_t n) { float* p = ws + off; off += n; return p; };
    float* tok_p   = alloc((size_t)N_TOK * D);   // later: h_srl_tok
    float* srl_p   = alloc((size_t)N_SRL * D);
    float* ent_p   = alloc((size_t)N_ENT * D);
    float* a_src_t = alloc((size_t)N_TOK * D);   // later: h1
    float* a_dst_s = alloc((size_t)N_SRL * D);
    float* a_dst_e = alloc((size_t)N_ENT * D);
    float* Pbuf    = alloc((size_t)(T_ROWS + 1) * D); // later: h_ent_tok
    float* csbuf   = alloc((size_t)NCH * D);
    float* big1    = alloc((size_t)E_EE * D);    // rel_emb -> tok_gh (50000*768)
    float* big2    = alloc((size_t)E_EE * D);    // m_ee    -> gi buffer
    float* big3    = alloc((size_t)E_EE * D);    // t_ee    -> gh buffer
    float* ent_r   = alloc((size_t)N_ENT * D);
    float* ent_rc  = alloc((size_t)N_ENT * D);
    float* Wrc     = alloc((size_t)D * D);
    float* umaxf   = alloc((size_t)N_SRL * D);   // shared TS then EE
    float* den     = alloc((size_t)N_SRL * D);
    float* srl_gi  = alloc((size_t)N_SRL * 3 * D);
    float* ent_gi  = alloc((size_t)N_ENT * 3 * D);
    unsigned* umax = (unsigned*)umaxf;
    (void)ws_size; (void)in_sizes; (void)n_in; (void)out_size;

    dim3 b256(256);

    // ---- node projections -------------------------------------------------
    gemm(feat_tok, N_TOK, D, W_nt, D, 0, b_nt, tok_p, D, stream);
    gemm(feat_srl, N_SRL, D, W_nt, D, 0, b_nt, srl_p, D, stream);
    gemm(feat_ent, N_ENT, D, W_nt, D, 0, b_nt, ent_p, D, stream);
    gemm(tok_p, N_TOK, D, W_att, 2 * D, 0, nullptr, a_src_t, D, stream);
    gemm(srl_p, N_SRL, D, W_att, 2 * D, D, b_att, a_dst_s, D, stream);
    gemm(ent_p, N_ENT, D, W_att, 2 * D, D, b_att, a_dst_e, D, stream);

    // ---- TS attention -> h_srl -------------------------------------------
    {
        int n = N_SRL * D;
        hipLaunchKernelGGL(fill_u32, dim3((n + 255) / 256), b256, 0, stream, umax, 0x007FFFFFu, n);
        hipLaunchKernelGGL(att_pass_max, dim3(E_TS), b256, 0, stream, a_src_t, src_ts, a_dst_s, dst_ts, umax, 0);
        hipLaunchKernelGGL(att_prep, dim3((n + 255) / 256), b256, 0, stream, umax, den, out_hsrl, n);
        hipLaunchKernelGGL(att_pass_den, dim3(E_TS), b256, 0, stream, a_src_t, src_ts, a_dst_s, dst_ts, umaxf, den, 0);
        hipLaunchKernelGGL(att_pass_agg, dim3(E_TS), b256, 0, stream, a_src_t, src_ts, a_dst_s, dst_ts, umaxf, den, tok_p, out_hsrl, 0);
    }

    // ---- span means -> rel_emb (big1) ------------------------------------
    hipLaunchKernelGGL(csum_partial, dim3(NCH), b256, 0, stream, bert, csbuf);
    hipLaunchKernelGGL(csum_scan, dim3(1), b256, 0, stream, csbuf);
    hipLaunchKernelGGL(csum_write, dim3(NCH), b256, 0, stream, bert, csbuf, Pbuf);
    hipLaunchKernelGGL(span_rel, dim3(E_EE), b256, 0, stream, Pbuf, span_idx, rel_type, big1);

    // ---- m_ee = ent_rc[src] + rel_emb @ Wrc^T ----------------------------
    hipLaunchKernelGGL(wrc_gemm, dim3(D), b256, 0, stream, W_com, W_rel, Wrc);
    gemm(feat_ent, N_ENT, D, W_rel, 2 * D, 0, b_rel, ent_r, D, stream);
    gemm(ent_r, N_ENT, D, W_com, D, 0, b_com, ent_rc, D, stream);
    gemm(big1, E_EE, D, Wrc, D, 0, nullptr, big2, D, stream);            // m_ee partial
    hipLaunchKernelGGL(edge_add_gather, dim3(E_EE), b256, 0, stream, big2, ent_rc, src_ee);
    gemm(big2, E_EE, D, W_att, 2 * D, 0, nullptr, big3, D, stream);      // t_ee

    // ---- EE attention -> h_ent -------------------------------------------
    {
        int n = N_ENT * D;
        hipLaunchKernelGGL(fill_u32, dim3((n + 255) / 256), b256, 0, stream, umax, 0x007FFFFFu, n);
        hipLaunchKernelGGL(att_pass_max, dim3(E_EE), b256, 0, stream, big3, src_ee, a_dst_e, dst_ee, umax, 1);
        hipLaunchKernelGGL(att_prep, dim3((n + 255) / 256), b256, 0, stream, umax, den, out_hent, n);
        hipLaunchKernelGGL(att_pass_den, dim3(E_EE), b256, 0, stream, big3, src_ee, a_dst_e, dst_ee, umaxf, den, 1);
        hipLaunchKernelGGL(att_pass_agg, dim3(E_EE), b256, 0, stream, big3, src_ee, a_dst_e, dst_ee, umaxf, den, big2, out_hent, 1);
    }

    // ---- GRU node-level precomputations ----------------------------------
    float* tok_gh = big1;   // rel_emb dead
    gemm(out_hsrl, N_SRL, D, W_ih, D, 0, b_ih, srl_gi, 3 * D, stream);
    gemm(out_hent, N_ENT, D, W_ih, D, 0, b_ih, ent_gi, 3 * D, stream);
    gemm(feat_tok, N_TOK, D, W_hh, D, 0, b_hh, tok_gh, 3 * D, stream);

    // ---- GRU edge messages + scatter-sum ---------------------------------
    float* h_srl_tok = tok_p;  // tok_p dead
    float* h_ent_tok = Pbuf;   // P dead
    {
        int n = N_TOK * D;
        hipLaunchKernelGGL(zero_kernel, dim3((n + 255) / 256), b256, 0, stream, h_srl_tok, n);
        hipLaunchKernelGGL(zero_kernel, dim3((n + 255) / 256), b256, 0, stream, h_ent_tok, n);
    }
    hipLaunchKernelGGL(gru_edge_agg, dim3(E_ST), b256, 0, stream, srl_gi, src_st, tok_gh, feat_tok, dst_st, h_srl_tok);
    hipLaunchKernelGGL(gru_edge_agg, dim3(E_ET), b256, 0, stream, ent_gi, src_et, tok_gh, feat_tok, dst_et, h_ent_tok);

    // ---- node GRUs: h1 then h_tok ----------------------------------------
    float* gi_buf = big2;  // m_ee dead
    float* gh_buf = big3;  // t_ee dead
    float* h1     = a_src_t;
    gemm(h_ent_tok, N_TOK, D, W_ih, D, 0, b_ih, gi_buf, 3 * D, stream);
    gemm(h_srl_tok, N_TOK, D, W_hh, D, 0, b_hh, gh_buf, 3 * D, stream);
    hipLaunchKernelGGL(gru_combine, dim3(N_TOK), b256, 0, stream, gi_buf, gh_buf, h_srl_tok, h1);

    gemm(feat_tok, N_TOK, D, W_ih, D, 0, b_ih, gi_buf, 3 * D, stream);
    gemm(h1, N_TOK, D, W_hh, D, 0, b_hh, gh_buf, 3 * D, stream);
    hipLaunchKernelGGL(gru_combine, dim3(N_TOK), b256, 0, stream, gi_buf, gh_buf, h1, out_htok);
}